// WrapperModule_44220983280307
// MI455X (gfx1250) — compile-verified
//
#include <hip/hip_runtime.h>
#include <hip/hip_bf16.h>
#include <stdint.h>

#define N_BOX 8192
#define CONF_TH 0.25f
#define IOU_TH 0.45f

typedef float v2f __attribute__((ext_vector_type(2)));
typedef float v8f __attribute__((ext_vector_type(8)));

__device__ __forceinline__ unsigned sortable_u32(float f) {
    // monotone map float -> unsigned (ascending)
    unsigned u = __float_as_uint(f);
    return u ^ ((u & 0x80000000u) ? 0xFFFFFFFFu : 0x80000000u);
}

// ---------------------------------------------------------------------------
// Kernel 1: decode + global coord max + bitonic sort by descending score +
// gather sorted boxes / scores / areas. One workgroup, 64KB LDS.
// ---------------------------------------------------------------------------
__global__ void __launch_bounds__(1024)
k_sort_gather(const float* __restrict__ preds,
              float4* __restrict__ boxes_s,
              float* __restrict__ scores_s,
              float* __restrict__ areas_s,
              unsigned* __restrict__ maxkey) {
    __shared__ unsigned long long keys[N_BOX];
    const int tid = threadIdx.x;

    unsigned lmax = 0u;
    for (int i = tid; i < N_BOX; i += 1024) {
        float cx = preds[0 * N_BOX + i];
        float cy = preds[1 * N_BOX + i];
        float w  = preds[2 * N_BOX + i];
        float h  = preds[3 * N_BOX + i];
        float ob = preds[4 * N_BOX + i];
        float x1 = cx - 0.5f * w, y1 = cy - 0.5f * h;
        float x2 = cx + 0.5f * w, y2 = cy + 0.5f * h;
        lmax = max(lmax, sortable_u32(x1));
        lmax = max(lmax, sortable_u32(y1));
        lmax = max(lmax, sortable_u32(x2));
        lmax = max(lmax, sortable_u32(y2));
        // ascending key == descending score, stable ties by ascending index
        unsigned s = sortable_u32(ob);
        keys[i] = ((unsigned long long)(~s) << 32) | (unsigned)i;
    }
    // wave32 reduction, then device atomic
    for (int off = 16; off > 0; off >>= 1)
        lmax = max(lmax, (unsigned)__shfl_xor((int)lmax, off, 32));
    if ((tid & 31) == 0) atomicMax(maxkey, lmax);
    __syncthreads();

    // bitonic sort (ascending) of 8192 u64 keys
    for (int k = 2; k <= N_BOX; k <<= 1) {
        for (int j = k >> 1; j > 0; j >>= 1) {
            for (int i = tid; i < N_BOX; i += 1024) {
                int ixj = i ^ j;
                if (ixj > i) {
                    unsigned long long a = keys[i];
                    unsigned long long b = keys[ixj];
                    bool up = ((i & k) == 0);
                    if (up ? (a > b) : (a < b)) {
                        keys[i] = b;
                        keys[ixj] = a;
                    }
                }
            }
            __syncthreads();
        }
    }

    // gather into sorted order
    for (int p = tid; p < N_BOX; p += 1024) {
        int idx = (int)(unsigned)keys[p];
        float cx = preds[0 * N_BOX + idx];
        float cy = preds[1 * N_BOX + idx];
        float w  = preds[2 * N_BOX + idx];
        float h  = preds[3 * N_BOX + idx];
        float ob = preds[4 * N_BOX + idx];
        float x1 = cx - 0.5f * w, y1 = cy - 0.5f * h;
        float x2 = cx + 0.5f * w, y2 = cy + 0.5f * h;
        boxes_s[p]  = make_float4(x1, y1, x2, y2);
        scores_s[p] = ob;
        areas_s[p]  = fmaxf(x2 - x1, 0.0f) * fmaxf(y2 - y1, 0.0f);
    }
}

// ---------------------------------------------------------------------------
// Kernel 2: pairwise IoU > thresh bitmask, 16x16 tiles per wave.
// union base (area_i + area_j) computed on the matrix pipe with
// V_WMMA_F32_16X16X4_F32 as a rank-2 outer sum; VALU does intersection.
// mask layout: row-major, 512 u16 halfwords per row (== 256 u32 words).
// Straight-line tile body: 8 ballots collected, then lanes 0..15 each emit
// their row's halfword via a cndmask select chain + one predicated store.
// ---------------------------------------------------------------------------
__global__ void __launch_bounds__(256)
k_mask(const float4* __restrict__ boxes_s,
       const float* __restrict__ areas_s,
       unsigned short* __restrict__ mask16) {
    const int lane = threadIdx.x & 31;
    const int hf   = lane >> 4;      // half-wave: 0 -> rows m, 1 -> rows m+8
    const int n    = lane & 15;      // column within tile
    const int wid  = blockIdx.x * (blockDim.x >> 5) + (threadIdx.x >> 5);
    const int nw   = gridDim.x * (blockDim.x >> 5);
    const int TILES = (N_BOX / 16) * (N_BOX / 16);

    for (int t = wid; t < TILES; t += nw) {
        const int ti = t >> 9;       // row tile  (512 tiles)
        const int tj = t & 511;      // col tile
        const int I = ti << 4;
        const int J = tj << 4;

        // A = [area_row | 1 | 0 | 0] (16x4), B = [1 ; area_col ; 0 ; 0] (4x16)
        // => D[m][n] = area[I+m] + area[J+n]
        v2f a = {0.0f, 0.0f};
        v2f b = {0.0f, 0.0f};
        if (hf == 0) {
            a = (v2f){areas_s[I + lane], 1.0f};
            b = (v2f){1.0f, areas_s[J + lane]};
        }
        v8f c = {0.f, 0.f, 0.f, 0.f, 0.f, 0.f, 0.f, 0.f};
        v8f d = __builtin_amdgcn_wmma_f32_16x16x4_f32(
            /*neg_a=*/false, a, /*neg_b=*/false, b,
            /*c_mod=*/(short)0, c, /*reuse_a=*/false, /*reuse_b=*/false);

        const float4 cb = boxes_s[J + n];

        unsigned bal[8];
#pragma unroll
        for (int r = 0; r < 8; ++r) {
            // element (m, n): m = I + r + 8*hf
            const float4 rb = boxes_s[I + r + (hf << 3)];
            float xx1 = fmaxf(rb.x, cb.x);
            float yy1 = fmaxf(rb.y, cb.y);
            float xx2 = fminf(rb.z, cb.z);
            float yy2 = fminf(rb.w, cb.w);
            float inter = fmaxf(xx2 - xx1, 0.0f) * fmaxf(yy2 - yy1, 0.0f);
            float un = d[r] - inter;                 // area_m + area_n - inter
            bool sup = (un > 0.0f) && (inter > IOU_TH * un);
            // low 16 bits: row I+r, cols J..J+15 ; high 16: row I+r+8
            bal[r] = __builtin_amdgcn_ballot_w32(sup);
        }

        // ballot words are wave-uniform: lane l (<16) owns row I+l.
        //   l < 8  -> low  half of bal[l]
        //   l >= 8 -> high half of bal[l-8]
        if (lane < 16) {
            const int r_ = lane & 7;
            unsigned v = bal[0];
#pragma unroll
            for (int q = 1; q < 8; ++q) v = (r_ == q) ? bal[q] : v;
            unsigned hv = (lane & 8) ? (v >> 16) : (v & 0xFFFFu);
            mask16[(unsigned)(I + lane) * 512u + tj] = (unsigned short)hv;
        }
    }
}

// ---------------------------------------------------------------------------
// Kernel 3: blocked sequential NMS scan over the bitmask + output.
// 256 blocks of 32 rows: wave0 resolves intra-block keeps with a shuffle
// loop, then all 1024 threads OR kept rows into the remv vector (LDS).
// ---------------------------------------------------------------------------
__global__ void __launch_bounds__(1024)
k_scan(const float4* __restrict__ boxes_s,
       const float* __restrict__ scores_s,
       const unsigned* __restrict__ mask32,
       const unsigned* __restrict__ maxkey,
       float* __restrict__ out) {
    __shared__ unsigned remv[256];   // suppressed bits, 1 bit per sorted index
    __shared__ unsigned keepw[256];  // keep bits
    const int tid = threadIdx.x;

    if (tid < 256) { remv[tid] = 0u; keepw[tid] = 0u; }
    __syncthreads();

    for (int blk = 0; blk < 256; ++blk) {
        // --- intra-block resolution (wave 0, 32 rows) ---
        if (tid < 32) {
            const int row = blk * 32 + tid;
            unsigned dr = mask32[(unsigned)row * 256u + blk]; // diagonal word
            bool val = scores_s[row] > CONF_TH;
            unsigned valmask = __builtin_amdgcn_ballot_w32(val);
            unsigned sup = remv[blk];
            unsigned keep = 0u;
#pragma unroll 1
            for (int k = 0; k < 32; ++k) {
                unsigned drk = (unsigned)__shfl((int)dr, k, 32);
                bool ki = ((valmask >> k) & 1u) && !((sup >> k) & 1u);
                if (ki) { keep |= (1u << k); sup |= drk; }
            }
            if (tid == 0) { keepw[blk] = keep; remv[blk] = sup; }
        }
        __syncthreads();

        // --- broadcast suppression of kept rows across all 256 words ---
        {
            const int w = tid & 255;
            const int g = tid >> 8;           // 4 groups split the 32 rows
            const unsigned k = keepw[blk];
            unsigned acc = 0u;
            for (int i = g; i < 32; i += 4) {
                if ((k >> i) & 1u)
                    acc |= mask32[(unsigned)(blk * 32 + i) * 256u + w];
            }
            if (acc) atomicOr(&remv[w], acc);
            if (blk + 1 < 256)  // warm L2/WGP$ for the next row block
                __builtin_prefetch(&mask32[(unsigned)((blk + 1) * 32 + g * 8) * 256u + w], 0, 1);
        }
        __syncthreads();
    }

    // --- output: scale = 416 iff max(boxes) <= 1.0 ; out coords = b*scale/416
    const float fac = (*maxkey <= 0xBF800000u /* sortable(1.0f) */)
                          ? 1.0f : (1.0f / 416.0f);
    for (int p = tid; p < N_BOX; p += 1024) {
        bool kp = (keepw[p >> 5] >> (p & 31)) & 1u;
        float4 bx = boxes_s[p];
        float s = scores_s[p];
        float* o = out + (size_t)p * 6;
        if (kp) {
            o[0] = bx.x * fac; o[1] = bx.y * fac;
            o[2] = bx.z * fac; o[3] = bx.w * fac;
            o[4] = s;          o[5] = 0.0f;
        } else {
            o[0] = 0.0f; o[1] = 0.0f; o[2] = 0.0f;
            o[3] = 0.0f; o[4] = 0.0f; o[5] = 0.0f;
        }
    }
}

// ---------------------------------------------------------------------------
// workspace layout (bytes):
//   [0,4)        maxkey (monotone-uint coord max)   -> padded to 256
//   [256, +128K) sorted boxes  float4[8192]
//   [+,   +32K)  sorted scores float[8192]
//   [+,   +32K)  sorted areas  float[8192]
//   [196864, +8MB) iou bitmask u16[8192*512] (aliased as u32[8192*256])
// ---------------------------------------------------------------------------
extern "C" void kernel_launch(void* const* d_in, const int* in_sizes, int n_in,
                              void* d_out, int out_size, void* d_ws, size_t ws_size,
                              hipStream_t stream) {
    (void)in_sizes; (void)n_in; (void)out_size; (void)ws_size;
    const float* preds = (const float*)d_in[0];
    float* out = (float*)d_out;
    char* ws = (char*)d_ws;

    unsigned* maxkey        = (unsigned*)ws;
    float4*   boxes_s       = (float4*)(ws + 256);
    float*    scores_s      = (float*)(ws + 256 + 131072);
    float*    areas_s       = (float*)(ws + 256 + 131072 + 32768);
    unsigned short* mask16  = (unsigned short*)(ws + 196864);

    hipMemsetAsync(maxkey, 0, sizeof(unsigned), stream);

    k_sort_gather<<<1, 1024, 0, stream>>>(preds, boxes_s, scores_s, areas_s, maxkey);
    k_mask<<<2048, 256, 0, stream>>>(boxes_s, areas_s, mask16);
    k_scan<<<1, 1024, 0, stream>>>(boxes_s, scores_s,
                                   (const unsigned*)mask16, maxkey, out);
}